// Quantization_8409545966132
// MI455X (gfx1250) — compile-verified
//
#include <hip/hip_runtime.h>
#include <hip/hip_bf16.h>

typedef __attribute__((ext_vector_type(2))) float v2f;
typedef __attribute__((ext_vector_type(8))) float v8f;

#define DDIM 256
#define KCODES 4096
#define NROWS 8192            // B*T = 16*512
#define ROWS_PER_BLOCK 64
#define RT 4                  // row tiles of 16 per block
#define COL_TILES (KCODES / 16)      // 256
#define HALF_TILES (COL_TILES / 2)   // 128 col tiles per column-half
#define WAVES 8
#define XSTRIDE 260           // padded LDS row stride (floats): conflict-free A reads

// ---------------- kernel 1: c2[k] = sum_d codebook[k][d]^2 ----------------
__global__ __launch_bounds__(256) void vq_c2_kernel(const float* __restrict__ cb,
                                                    float* __restrict__ c2) {
    int k = blockIdx.x * blockDim.x + threadIdx.x;            // 0..4095 exact
    const float4* row = (const float4*)(cb + (size_t)k * DDIM);
    float s = 0.f;
#pragma unroll
    for (int i = 0; i < DDIM / 4; ++i) {
        float4 v = row[i];
        s += v.x * v.x + v.y * v.y + v.z * v.z + v.w * v.w;
    }
    c2[k] = s;
}

// ---------- kernel 2: WMMA f32 GEMM + per-column-half argmax --------------
// grid = (NROWS/64 row blocks) x (2 column halves) = 256 blocks
__global__ __launch_bounds__(256) void vq_argmax_kernel(const float* __restrict__ x,
                                                        const float* __restrict__ cb,
                                                        const float* __restrict__ c2,
                                                        float* __restrict__ candV,
                                                        int* __restrict__ candI) {
    __shared__ float smem[ROWS_PER_BLOCK * XSTRIDE];   // 65 KB, reused for reduction

    const int tid     = threadIdx.x;
    const int wave    = tid >> 5;
    const int lane    = tid & 31;
    const int r       = lane & 15;        // row-in-tile for A, col-in-tile for B/C
    const int half    = lane >> 4;        // 0 or 1
    const int khalf   = half * 2;         // K offset inside each 4-wide WMMA step
    const int colHalf = blockIdx.x & 1;   // which half of the codebook
    const int rowBlk  = blockIdx.x >> 1;
    const size_t rowBase = (size_t)rowBlk * ROWS_PER_BLOCK;

    // ---- stage 64x256 X tile into LDS (padded row stride) ----
    {
        const int row = tid >> 2;         // 0..63  (4 threads per row)
        const int seg = tid & 3;          // 4 segs * 64 floats
        const float4* src = (const float4*)(x + (rowBase + row) * DDIM + seg * 64);
        float4* dst = (float4*)(smem + row * XSTRIDE + seg * 64);
#pragma unroll
        for (int j = 0; j < 16; ++j) dst[j] = src[j];
    }
    __syncthreads();

    float bestV[RT][8];
    int   bestI[RT][8];
#pragma unroll
    for (int t = 0; t < RT; ++t)
#pragma unroll
        for (int v = 0; v < 8; ++v) { bestV[t][v] = -3.402823466e38f; bestI[t][v] = 0; }

    // ---- scan this half's 128 column tiles, 8 waves interleaved ----
    const int ctBeg = colHalf * HALF_TILES;
    const int ctEnd = ctBeg + HALF_TILES;
    for (int ct = ctBeg + wave; ct < ctEnd; ct += WAVES) {
        const int col = (ct << 4) | r;                       // this lane's code index
        const float* __restrict__ bp = cb + (size_t)col * DDIM;

        // prefetch next tile's B rows (covers full 16KB tile at 128B strides)
        if (ct + WAVES < ctEnd) {
            const float* np = cb + (size_t)(((ct + WAVES) << 4) | r) * DDIM;
#pragma unroll
            for (int p = 0; p < 8; ++p) __builtin_prefetch(np + p * 32, 0, 1);
        }

        v8f acc[RT] = {};
#pragma unroll 8
        for (int kk = 0; kk < DDIM / 4; ++kk) {
            const int kb = kk * 4 + khalf;
            // B frag 4x16: lane=N holds K={kb, kb+1} -> contiguous b64 global load
            v2f b = *(const v2f*)(bp + kb);
#pragma unroll
            for (int t = 0; t < RT; ++t) {
                // A frag 16x4: lane=M holds K={kb, kb+1} -> contiguous ds_load_b64
                v2f a = *(const v2f*)(smem + (t * 16 + r) * XSTRIDE + kb);
                acc[t] = __builtin_amdgcn_wmma_f32_16x16x4_f32(
                    false, a, false, b, (short)0, acc[t], false, false);
            }
        }

        const float cc = c2[col];
#pragma unroll
        for (int t = 0; t < RT; ++t)
#pragma unroll
            for (int v = 0; v < 8; ++v) {
                float s = __builtin_fmaf(-2.0f, acc[t][v], cc);  // c2 - 2*dot
                if (s > bestV[t][v]) { bestV[t][v] = s; bestI[t][v] = col; }
            }
    }
    __syncthreads();        // X tile no longer needed; alias smem for reduction

    // ---- per-row reduction over 8 waves x 16 lanes = 128 candidates ----
    float* rv = smem;                                        // [64][128] floats (32 KB)
    int*   ri = (int*)(smem + ROWS_PER_BLOCK * 128);         // [64][128] ints   (32 KB)
    const int slot = wave * 16 + r;                          // 0..127
#pragma unroll
    for (int t = 0; t < RT; ++t)
#pragma unroll
        for (int v = 0; v < 8; ++v) {
            int row = t * 16 + half * 8 + v;                 // C/D layout: M = v (+8 half 1)
            rv[row * 128 + slot] = bestV[t][v];
            ri[row * 128 + slot] = bestI[t][v];
        }
    __syncthreads();

    if (tid < ROWS_PER_BLOCK) {
        float bv = -3.402823466e38f; int bi = KCODES;
        for (int s = 0; s < 128; ++s) {
            float v = rv[tid * 128 + s];
            int   i = ri[tid * 128 + s];
            if (v > bv || (v == bv && i < bi)) { bv = v; bi = i; }  // first-occurrence
        }
        const size_t row = rowBase + tid;
        candV[row * 2 + colHalf] = bv;
        candI[row * 2 + colHalf] = bi;
    }
}

// ---------- kernel 3: pick winner of the two halves, gather codebook row ----
__global__ __launch_bounds__(256) void vq_gather_kernel(const float* __restrict__ cb,
                                                        const float* __restrict__ candV,
                                                        const int* __restrict__ candI,
                                                        float* __restrict__ out) {
    const size_t row = (size_t)blockIdx.x * 32 + (threadIdx.x >> 3);
    const int seg = threadIdx.x & 7;          // 8 segs * 32 floats
    const float v0 = candV[row * 2 + 0];
    const float v1 = candV[row * 2 + 1];
    const int   i0 = candI[row * 2 + 0];
    const int   i1 = candI[row * 2 + 1];
    // half-0 indices are all smaller, so ties keep i0 = first occurrence
    const int idx = (v1 > v0) ? i1 : i0;
    const float4* src = (const float4*)(cb + (size_t)idx * DDIM + seg * 32);
    float4* dst = (float4*)(out + row * DDIM + seg * 32);
#pragma unroll
    for (int j = 0; j < 8; ++j) dst[j] = src[j];
}

extern "C" void kernel_launch(void* const* d_in, const int* in_sizes, int n_in,
                              void* d_out, int out_size, void* d_ws, size_t ws_size,
                              hipStream_t stream) {
    const float* x  = (const float*)d_in[0];   // [8192, 256]
    const float* cb = (const float*)d_in[1];   // [4096, 256]
    float* out = (float*)d_out;                // [8192, 256]

    // workspace layout: c2[4096] | candV[8192*2] | candI[8192*2]  (~144 KB)
    float* c2    = (float*)d_ws;
    float* candV = c2 + KCODES;
    int*   candI = (int*)(candV + NROWS * 2);

    vq_c2_kernel<<<KCODES / 256, 256, 0, stream>>>(cb, c2);
    vq_argmax_kernel<<<(NROWS / ROWS_PER_BLOCK) * 2, 256, 0, stream>>>(x, cb, c2, candV, candI);
    vq_gather_kernel<<<NROWS / 32, 256, 0, stream>>>(cb, candV, candI, out);
}